// noflayer_38233798869431
// MI455X (gfx1250) — compile-verified
//
#include <hip/hip_runtime.h>
#include <hip/hip_bf16.h>

// ---------------------------------------------------------------------------
// NOF/GAT-style propagation layer for MI455X (gfx1250, wave32).
//  - Dense projection x@a uses V_WMMA_F32_16X16X4_F32 (one wave per 16-row tile)
//  - Edge softmax + SpMM hops are edge-parallel with native f32 L2 atomics
//    (source matrix [N,128] f32 = 51.2MB fits the 192MB L2 -> gathers hit L2)
// ---------------------------------------------------------------------------

#define ALPHA_ 0.1f
#define LEAKY_ 0.2f
#define FDIM   128

typedef __attribute__((ext_vector_type(2))) float v2f;
typedef __attribute__((ext_vector_type(8))) float v8f;

// Order-preserving float<->uint transform for atomicMax on floats.
__device__ __forceinline__ unsigned f2key(float f) {
    unsigned u = __float_as_uint(f);
    return (u & 0x80000000u) ? ~u : (u | 0x80000000u);
}
__device__ __forceinline__ float key2f(unsigned k) {
    return (k & 0x80000000u) ? __uint_as_float(k & 0x7fffffffu)
                             : __uint_as_float(~k);
}
__device__ __forceinline__ void atomAddF(float* p, float v) {
    // relaxed agent-scope fadd -> global_atomic_add_f32 (no CAS loop)
    __hip_atomic_fetch_add(p, v, __ATOMIC_RELAXED, __HIP_MEMORY_SCOPE_AGENT);
}
__device__ __forceinline__ float sigmoidf_(float v) {
    return 1.0f / (1.0f + __expf(-v));
}

// ---------------------------------------------------------------------------
// Phase 1: feat1 = x @ a[0:128], feat2 = x @ a[128:256] via WMMA f32 16x16x4.
// One wave handles a 16-row tile of x. A-matrix layout (ISA 7.12.2, 32-bit A
// 16x4): lanes 0-15 -> M=lane, {v0,v1} = K{0,1}; lanes 16-31 -> M=lane-16,
// {v0,v1} = K{2,3}. B (4x16): lanes 0-15 -> N=lane, {v0,v1} = K{0,1};
// lanes 16-31 -> N=lane-16, {v0,v1} = K{2,3}. C/D 16x16: VGPR j, lanes 0-15:
// (M=j, N=lane); lanes 16-31: (M=j+8, N=lane-16).
//
// B loads are unconditional (all lanes read in-bounds bytes of the 256-float
// `a` vector) and masked with a multiply, so EXEC stays all-ones through the
// whole loop (WMMA requirement) and no saveexec/branch churn is emitted.
// ---------------------------------------------------------------------------
__global__ void k_feat_wmma(const float* __restrict__ x,
                            const float* __restrict__ a,
                            float* __restrict__ f1,
                            float* __restrict__ f2, int N) {
    int wave  = (int)((blockIdx.x * (unsigned)blockDim.x + threadIdx.x) >> 5);
    int lane  = threadIdx.x & 31;
    int tiles = (N + 15) >> 4;
    if (wave >= tiles) return;               // uniform per wave: EXEC stays full

    int m    = lane & 15;
    int koff = (lane >> 4) << 1;             // 0 for lanes 0-15, 2 for 16-31
    int row  = wave * 16 + m;
    int rowc = row < N ? row : N - 1;        // clamp tail loads (stores guarded)
    const float* xr = x + (long long)rowc * FDIM;

    // Lane's B column: n = m. Columns 0/1 carry a1/a2; others contribute 0.
    const float* ab  = a + ((m < 2) ? (m * FDIM) : 0);  // in-bounds for all k
    const float bmsk = (m < 2) ? 1.f : 0.f;

    v8f c = {0.f, 0.f, 0.f, 0.f, 0.f, 0.f, 0.f, 0.f};
    #pragma unroll
    for (int k = 0; k < FDIM; k += 4) {
        v2f av, bv;
        av.x = xr[k + koff];
        av.y = xr[k + koff + 1];
        bv.x = ab[k + koff] * bmsk;
        bv.y = ab[k + koff + 1] * bmsk;
        c = __builtin_amdgcn_wmma_f32_16x16x4_f32(
                /*neg_a=*/false, av, /*neg_b=*/false, bv,
                /*c_mod=*/(short)0, c, /*reuse_a=*/false, /*reuse_b=*/false);
    }

    int mbase = wave * 16 + ((lane >> 4) ? 8 : 0);
    if (m == 0) {
        #pragma unroll
        for (int j = 0; j < 8; ++j) {
            int r = mbase + j;
            if (r < N) f1[r] = c[j];
        }
    } else if (m == 1) {
        #pragma unroll
        for (int j = 0; j < 8; ++j) {
            int r = mbase + j;
            if (r < N) f2[r] = c[j];
        }
    }
}

// ---------------------------------------------------------------------------
// Node-state init: rmax keys to 0 (< key of every float incl. -inf),
// denom/rowsum to 0.
// ---------------------------------------------------------------------------
__global__ void k_init_nodes(unsigned* __restrict__ rmax,
                             float* __restrict__ denom,
                             float* __restrict__ rowsum, int N) {
    int i = blockIdx.x * blockDim.x + threadIdx.x;
    if (i >= N) return;
    rmax[i]   = 0u;
    denom[i]  = 0.f;
    rowsum[i] = 0.f;
}

// e = leakyrelu(f1[row] + f2[col]); store e; segment-max via uint atomicMax.
__global__ void k_edge_e(const float* __restrict__ f1,
                         const float* __restrict__ f2,
                         const int* __restrict__ row,
                         const int* __restrict__ col,
                         float* __restrict__ evals,
                         unsigned* __restrict__ rmax, int E) {
    int i = blockIdx.x * blockDim.x + threadIdx.x;
    if (i >= E) return;
    int r = row[i];
    float e = f1[r] + f2[col[i]];
    e = (e >= 0.f) ? e : LEAKY_ * e;
    evals[i] = e;
    atomicMax(&rmax[r], f2key(e));
}

// ex = exp(e - max[row]); store; segment-sum the denominator.
__global__ void k_edge_exp(float* __restrict__ evals,
                           const int* __restrict__ row,
                           const unsigned* __restrict__ rmax,
                           float* __restrict__ denom, int E) {
    int i = blockIdx.x * blockDim.x + threadIdx.x;
    if (i >= E) return;
    int r = row[i];
    float ex = __expf(evals[i] - key2f(rmax[r]));
    evals[i] = ex;
    atomAddF(&denom[r], ex);
}

// U = ex / denom[row]; rowsum[row] += adj * 0.5 * U  (P = 0.5*U folded in).
__global__ void k_edge_norm(float* __restrict__ evals,
                            const int* __restrict__ row,
                            const float* __restrict__ denom,
                            const float* __restrict__ adj,
                            float* __restrict__ rowsum, int E) {
    int i = blockIdx.x * blockDim.x + threadIdx.x;
    if (i >= E) return;
    int r = row[i];
    float u = evals[i] / denom[r];
    evals[i] = u;                               // evals now holds U
    atomAddF(&rowsum[r], adj[i] * 0.5f * u);
}

// Zero a float buffer (float4 vectorized).
__global__ void k_zero_f4(float4* __restrict__ p, long long n4) {
    long long i = (long long)blockIdx.x * blockDim.x + threadIdx.x;
    if (i < n4) p[i] = make_float4(0.f, 0.f, 0.f, 0.f);
}

// SpMM scatter: one wave per edge; lane handles 4 consecutive features.
// Gather of src rows is L2-resident (51.2MB < 192MB L2).
__global__ void k_spmm(const float* __restrict__ src,
                       float* __restrict__ dst,
                       const float* __restrict__ U,
                       const int* __restrict__ row,
                       const int* __restrict__ col, int E) {
    long long t = (long long)blockIdx.x * blockDim.x + threadIdx.x;
    int edge = (int)(t >> 5);
    if (edge >= E) return;
    int lane = (int)(t & 31);
    float u = U[edge];
    long long c = (long long)col[edge] * FDIM + lane * 4;
    long long r = (long long)row[edge] * FDIM + lane * 4;
    float4 v = *(const float4*)(src + c);
    float* d = dst + r;
    atomAddF(d + 0, u * v.x);
    atomAddF(d + 1, u * v.y);
    atomAddF(d + 2, u * v.z);
    atomAddF(d + 3, u * v.w);
}

// Fuse: feat_fuse = upd*(A + (1-A)*(1 - rowsum*cscale));
// hop0: out = A*x + (1-A)*fuse ; else: out = A*out + (1-A)*fuse.
// cheb scaling is folded into cscale so rowsum is never mutated.
__global__ void k_fuse(const float* __restrict__ upd,
                       const float* __restrict__ x,
                       const float* __restrict__ rowsum,
                       const float* __restrict__ cheb,
                       float* __restrict__ out, int hop, int N) {
    long long t = (long long)blockIdx.x * blockDim.x + threadIdx.x;
    int node = (int)(t >> 5);
    if (node >= N) return;
    int lane = (int)(t & 31) * 4;

    float cs = 1.f;
    if (hop >= 1) cs *= sigmoidf_(cheb[0]);
    if (hop >= 2) cs *= sigmoidf_(cheb[1]);
    float rs = rowsum[node] * cs;
    float s  = ALPHA_ + (1.f - ALPHA_) * (1.f - rs);

    long long off = (long long)node * FDIM + lane;
    float4 u = *(const float4*)(upd + off);
    float4 o;
    if (hop == 0) {
        float4 xv = *(const float4*)(x + off);
        o.x = ALPHA_ * xv.x + (1.f - ALPHA_) * (s * u.x);
        o.y = ALPHA_ * xv.y + (1.f - ALPHA_) * (s * u.y);
        o.z = ALPHA_ * xv.z + (1.f - ALPHA_) * (s * u.z);
        o.w = ALPHA_ * xv.w + (1.f - ALPHA_) * (s * u.w);
    } else {
        float4 p = *(const float4*)(out + off);
        o.x = ALPHA_ * p.x + (1.f - ALPHA_) * (s * u.x);
        o.y = ALPHA_ * p.y + (1.f - ALPHA_) * (s * u.y);
        o.z = ALPHA_ * p.z + (1.f - ALPHA_) * (s * u.z);
        o.w = ALPHA_ * p.w + (1.f - ALPHA_) * (s * u.w);
    }
    *(float4*)(out + off) = o;
}

// ---------------------------------------------------------------------------
extern "C" void kernel_launch(void* const* d_in, const int* in_sizes, int n_in,
                              void* d_out, int out_size, void* d_ws, size_t ws_size,
                              hipStream_t stream) {
    const float* x    = (const float*)d_in[0];
    // d_in[1] (h0) and d_in[3] (temp) are unused by the reference.
    const float* a    = (const float*)d_in[2];
    const float* cheb = (const float*)d_in[4];
    const float* adj  = (const float*)d_in[5];
    const int*   erow = (const int*)d_in[6];
    const int*   ecol = (const int*)d_in[7];

    const int N = in_sizes[0] / FDIM;
    const int E = in_sizes[5];

    // Workspace carve-up (needs ~2*N*F*4 + E*4 + 5*N*4 bytes ~ 110MB).
    char* w = (char*)d_ws;
    auto carve = [&](size_t bytes) -> char* {
        char* p = w;
        w += (bytes + 255) & ~(size_t)255;
        return p;
    };
    float*    feat1  = (float*)   carve((size_t)N * 4);
    float*    feat2  = (float*)   carve((size_t)N * 4);
    unsigned* rmax   = (unsigned*)carve((size_t)N * 4);
    float*    denom  = (float*)   carve((size_t)N * 4);
    float*    rowsum = (float*)   carve((size_t)N * 4);
    float*    U      = (float*)   carve((size_t)E * 4);
    float*    updA   = (float*)   carve((size_t)N * FDIM * 4);
    float*    updB   = (float*)   carve((size_t)N * FDIM * 4);
    float*    out    = (float*)d_out;

    // 1) Projection (WMMA): 8 waves/block, 1 tile (16 rows) per wave.
    {
        int tiles  = (N + 15) >> 4;
        int blocks = (tiles + 7) / 8;
        k_feat_wmma<<<blocks, 256, 0, stream>>>(x, a, feat1, feat2, N);
    }
    // 2) Node state init.
    k_init_nodes<<<(N + 255) / 256, 256, 0, stream>>>(rmax, denom, rowsum, N);
    // 3) Edge softmax (3 passes).
    int eblk = (E + 255) / 256;
    k_edge_e   <<<eblk, 256, 0, stream>>>(feat1, feat2, erow, ecol, U, rmax, E);
    k_edge_exp <<<eblk, 256, 0, stream>>>(U, erow, rmax, denom, E);
    k_edge_norm<<<eblk, 256, 0, stream>>>(U, erow, denom, adj, rowsum, E);

    // 4) Three propagation hops.
    long long nf4   = (long long)N * FDIM / 4;
    int zblk        = (int)((nf4 + 255) / 256);
    long long et    = (long long)E * 32;
    int sblk        = (int)((et + 255) / 256);
    long long nt    = (long long)N * 32;
    int fblk        = (int)((nt + 255) / 256);

    const float* cur = x;           // update starts as x
    float* bufs[2] = {updA, updB};
    for (int hop = 0; hop < 3; ++hop) {
        float* nxt = bufs[hop & 1];
        k_zero_f4<<<zblk, 256, 0, stream>>>((float4*)nxt, nf4);
        k_spmm   <<<sblk, 256, 0, stream>>>(cur, nxt, U, erow, ecol, E);
        k_fuse   <<<fblk, 256, 0, stream>>>(nxt, x, rowsum, cheb, out, hop, N);
        cur = nxt;
    }
}